// CoupledFUSERoute_58059367907622
// MI455X (gfx1250) — compile-verified
//
#include <hip/hip_runtime.h>
#include <cstdint>

// =============================================================================
// Muskingum-Cunge tree routing, MI455X (gfx1250).
//   - 4 workgroups (1 WGP each), each owns one 2047-node subtree rooted at
//     global heap nodes 3..6; wg0 additionally serializes top nodes 0,1,2.
//   - Node->thread mapping is fixed across timesteps, so I_old/O_old state and
//     all leaf-level parameters live in REGISTERS; LDS holds only the outflow
//     vector (parent<-child exchange), non-leaf parameters, and the inflow
//     double buffer.
//   - Lateral-inflow rows (32KB/step) streamed HBM->LDS with the Tensor Data
//     Mover (double buffered, s_wait_tensorcnt fenced).
//   - Top 5 subtree levels processed by a single wave32 (in-order LDS pipe),
//     cutting s_barrier count per timestep.
//   - Cross-WG sync: acquire/release flags in d_ws (+ cluster barrier when
//     the dispatch forms a cluster).
// =============================================================================

#define RDEPTH    13
#define NREACH    8191          // 2^13 - 1
#define TSTEPS    2048
#define NWG       4             // one subtree per WGP
#define TPB       1024          // 32 wave32s
#define LATSTRIDE 8192
#define NPARAM    1024          // non-leaf nodes per subtree (j = 0..1022)

#if defined(__has_builtin)
# if __has_builtin(__builtin_amdgcn_tensor_load_to_lds)
#  define HAVE_TDM 1
# endif
# if __has_builtin(__builtin_amdgcn_s_cluster_barrier)
#  define HAVE_CBAR 1
# endif
# if __has_builtin(__builtin_amdgcn_groupstaticsize)
#  define HAVE_GSS 1
# endif
#endif
#ifndef HAVE_TDM
# define HAVE_TDM 0
#endif
#ifndef HAVE_CBAR
# define HAVE_CBAR 0
#endif
#ifndef HAVE_GSS
# define HAVE_GSS 0
#endif

#if __has_include(<hip/amd_detail/amd_gfx1250_TDM.h>)
# define TDM_6ARG 1               // therock-10.0 headers => 6-arg builtin
#else
# define TDM_6ARG 0               // ROCm 7.2 => 5-arg builtin
#endif

// ---- fast transcendental helpers (v_exp_f32 / v_log_f32 when available) ----
__device__ __forceinline__ float fast_exp2(float x) {
#if defined(__has_builtin) && __has_builtin(__builtin_amdgcn_exp2f)
  return __builtin_amdgcn_exp2f(x);
#else
  return exp2f(x);
#endif
}
__device__ __forceinline__ float fast_log2(float x) {
#if defined(__has_builtin) && __has_builtin(__builtin_amdgcn_logf)
  return __builtin_amdgcn_logf(x);   // v_log_f32 == log2(x)
#else
  return log2f(x);
#endif
}

#if HAVE_TDM
typedef unsigned int u32x4 __attribute__((ext_vector_type(4)));
typedef int          i32x4 __attribute__((ext_vector_type(4)));
typedef int          i32x8 __attribute__((ext_vector_type(8)));

// One-row (1-D) TDM copy: nelem fp32 values, global -> LDS (byte offset).
__device__ __forceinline__ void tdm_load_row(const float* gsrc,
                                             unsigned lds_byte,
                                             unsigned nelem) {
  unsigned long long ga = (unsigned long long)(uintptr_t)gsrc;
  u32x4 g0 = {0u, 0u, 0u, 0u};
  g0[0] = 1u;                                        // D# count=1, user mode
  g0[1] = lds_byte;                                  // lds_addr (bytes)
  g0[2] = (unsigned)ga;                              // global_addr[31:0]
  g0[3] = (unsigned)((ga >> 32) & 0x01FFFFFFull)     // global_addr[56:32]
        | (2u << 30);                                // type = 2 ("image")
  i32x8 g1 = {0, 0, 0, 0, 0, 0, 0, 0};
  g1[0] = (int)(2u << 16);                           // data_size=4B, mask=0
  g1[1] = (int)((nelem & 0xFFFFu) << 16);            // tensor_dim0[15:0]
  g1[2] = (int)((nelem >> 16) | (1u << 16));         // dim0[31:16] | tensor_dim1=1
  g1[3] = (int)((nelem & 0xFFFFu) << 16);            // tile_dim0
  g1[4] = 1;                                         // tile_dim1 = 1 row
  g1[5] = (int)nelem;                                // tensor_dim0_stride
  i32x4 z4 = {0, 0, 0, 0};
#if TDM_6ARG
  i32x8 z8 = {0, 0, 0, 0, 0, 0, 0, 0};
  __builtin_amdgcn_tensor_load_to_lds(g0, g1, z4, z4, z8, 0);
#else
  __builtin_amdgcn_tensor_load_to_lds(g0, g1, z4, z4, 0);
#endif
}
#endif  // HAVE_TDM

// Fused per-reach constants:
//   c  = max(Ap * Qr^E1, 1e-3)           (Ap = 5/3 * dc^(2/3) * sqrt(S)/n, E1 = 2de/3)
//   Qr/(w*S*c*L) = B * Qr^E2 / c         (B  = 1/(wc*S*L),                E2 = 1-we)
struct MCP { float Ap, E1, L, B, E2; };

__device__ __forceinline__ MCP make_p(float n, float L, float S, float wc,
                                      float we, float dc, float de) {
  MCP p;
  const float k23 = 0.6666666667f;
  float A = fast_exp2(k23 * fast_log2(dc)) * sqrtf(S) / n;
  p.Ap = 1.6666666667f * A;
  p.E1 = k23 * de;
  p.L  = L;
  p.B  = 1.0f / (wc * S * L);
  p.E2 = 1.0f - we;
  return p;
}

__device__ __forceinline__ float mc_step(const MCP& p, float I_new,
                                         float& I_old, float& O_old, float dtf) {
  float Qr  = fmaxf(0.5f * (I_new + O_old), 1e-3f);
  float lq  = fast_log2(Qr);                     // one log feeds both powers
  float c   = fmaxf(p.Ap * fast_exp2(p.E1 * lq), 1e-3f);
  float rc  = 1.0f / c;
  float K   = p.L * rc;
  float X   = 0.5f * (1.0f - p.B * fast_exp2(p.E2 * lq) * rc);
  X         = fminf(fmaxf(X, 0.0f), 0.5f);
  float KX  = 2.0f * K * X;
  float KmX = 2.0f * K - KX;                     // 2K(1-X)
  float rdn = 1.0f / (KmX + dtf);
  float O   = fmaxf(((dtf - KX) * I_new + (dtf + KX) * I_old +
                     (KmX - dtf) * O_old) * rdn, 0.0f);
  I_old = I_new;
  O_old = O;
  return O;
}

__global__ void mc_zero_ws(int* ws) {
  if (threadIdx.x < 16) ws[threadIdx.x] = 0;
}

__global__ void __launch_bounds__(TPB)
mc_route_kernel(const float* __restrict__ lat_g,
                const float* __restrict__ mn,  const float* __restrict__ len,
                const float* __restrict__ slp, const float* __restrict__ wcf,
                const float* __restrict__ wex, const float* __restrict__ dcf,
                const float* __restrict__ dex, const int* __restrict__ dtp,
                float* __restrict__ out, int* __restrict__ ws) {
  extern __shared__ float smem[];
  float* latb = smem;                       // [2][LATSTRIDE] inflow double buffer
  float* Qs   = smem + 2 * LATSTRIDE;       // [2048] outflows (child->parent)
  float* pAp  = Qs  + 2048;                 // fused params, non-leaf j = 0..1022
  float* pE1  = pAp + NPARAM;
  float* pL   = pE1 + NPARAM;
  float* pB   = pL  + NPARAM;
  float* pE2  = pB  + NPARAM;

  const int   tid = (int)threadIdx.x;
  const int   k   = (int)blockIdx.x;        // subtree id (global root node 3+k)
  const int   wv  = tid >> 5;               // wave32 index
  const float dtf = (float)dtp[0];

  int*   flags = ws;                        // [4]  subtree-root publish counters
  int*   gdone = ws + 4;                    // [1]  root consumption counter
  float* qslot = reinterpret_cast<float*>(ws + 8);   // [2][4] subtree-root Q

  // ---- init: zero outflow vector, precompute fused parameters ----
  for (int j = tid; j < 2048; j += TPB) Qs[j] = 0.0f;
  if (tid < 1023) {                          // non-leaf params -> LDS
    int j = tid;
    int m = 31 - __clz(j + 1);               // local level 0..9
    int i = j - ((1 << m) - 1);
    int g = ((1 << (m + 2)) - 1) + (k << m) + i;
    MCP p = make_p(mn[g], len[g], slp[g], wcf[g], wex[g], dcf[g], dex[g]);
    pAp[j] = p.Ap; pE1[j] = p.E1; pL[j] = p.L; pB[j] = p.B; pE2[j] = p.E2;
  }
  // leaf node (m=10) params + state -> registers (1 leaf per thread)
  const int gleaf = 4095 + (k << 10) + tid;  // global heap index of my leaf
  const MCP pleaf = make_p(mn[gleaf], len[gleaf], slp[gleaf], wcf[gleaf],
                           wex[gleaf], dcf[gleaf], dex[gleaf]);
  float leafIo = 0.0f, leafOo = 0.0f;

  // register state for owned nodes of levels m = 9..5 (index 9-m)
  float mIo[5] = {0.f, 0.f, 0.f, 0.f, 0.f};
  float mOo[5] = {0.f, 0.f, 0.f, 0.f, 0.f};
  // register state for wave-0 levels m = 4..0 (index 4-m, used when tid < 2^m)
  float wIo[5] = {0.f, 0.f, 0.f, 0.f, 0.f};
  float wOo[5] = {0.f, 0.f, 0.f, 0.f, 0.f};

  // top-of-tree nodes 0..2: state in wg0/thread0 registers
  MCP p0{}, p1{}, p2{};
  float I0o = 0.f, O0o = 0.f, I1o = 0.f, O1o = 0.f, I2o = 0.f, O2o = 0.f;
  if (k == 0 && tid == 0) {
    p0 = make_p(mn[0], len[0], slp[0], wcf[0], wex[0], dcf[0], dex[0]);
    p1 = make_p(mn[1], len[1], slp[1], wcf[1], wex[1], dcf[1], dex[1]);
    p2 = make_p(mn[2], len[2], slp[2], wcf[2], wex[2], dcf[2], dex[2]);
  }

#if HAVE_TDM
# if HAVE_GSS
  const unsigned lds0 = (unsigned)__builtin_amdgcn_groupstaticsize();
# else
  const unsigned lds0 = 0u;
# endif
  if (wv == 0) {                                   // preload inflow row 0
    tdm_load_row(lat_g, lds0, NREACH);
    __builtin_amdgcn_s_wait_tensorcnt(0);
  }
#endif
  __syncthreads();

  for (int t = 0; t < TSTEPS; ++t) {
#if HAVE_TDM
    const float* lat = latb + (t & 1) * LATSTRIDE;
    if (wv == 0 && (t + 1) < TSTEPS)               // async-prefetch next row
      tdm_load_row(lat_g + (size_t)(t + 1) * NREACH,
                   lds0 + (unsigned)(((t + 1) & 1) * LATSTRIDE * 4), NREACH);
#else
    const float* lat = lat_g + (size_t)t * NREACH;
#endif

    // ---- leaf level (m=10): all state/params in registers ----
    {
      const float O = mc_step(pleaf, lat[gleaf], leafIo, leafOo, dtf);
      Qs[1023 + tid] = O;                          // publish for parent
    }
    __syncthreads();

    // ---- levels m = 9..5 : whole workgroup, barrier per level ----
#pragma unroll
    for (int m = 9; m >= 5; --m) {
      const int cnt = 1 << m;
      if (tid < cnt) {
        const int   j  = cnt - 1 + tid;
        const float up = Qs[2 * j + 1] + Qs[2 * j + 2];
        const int   g  = ((1 << (m + 2)) - 1) + (k << m) + tid;
        MCP p{pAp[j], pE1[j], pL[j], pB[j], pE2[j]};
        const float O = mc_step(p, lat[g] + up, mIo[9 - m], mOo[9 - m], dtf);
        Qs[j] = O;                                 // publish for parent
      }
      __syncthreads();
    }

    // ---- levels m = 4..0 : single wave32 (in-order LDS pipe) ----
    if (wv == 0) {
      float qsub = 0.0f;
#pragma unroll
      for (int m = 4; m >= 0; --m) {
        const int cnt = 1 << m;
        if (tid < cnt) {
          const int   j  = cnt - 1 + tid;
          const float up = Qs[2 * j + 1] + Qs[2 * j + 2];
          const int   g  = ((1 << (m + 2)) - 1) + (k << m) + tid;
          MCP p{pAp[j], pE1[j], pL[j], pB[j], pE2[j]};
          const float O = mc_step(p, lat[g] + up, wIo[4 - m], wOo[4 - m], dtf);
          Qs[j] = O;
          if (m == 0) qsub = O;
        }
        __asm__ __volatile__("" ::: "memory");   // keep cross-lane LDS order
      }

      if (tid == 0) {
        if (k != 0) {                            // publish subtree-root Q
          __hip_atomic_store(&qslot[(t & 1) * NWG + k], qsub,
                             __ATOMIC_RELAXED, __HIP_MEMORY_SCOPE_AGENT);
          __hip_atomic_store(&flags[k], t + 1,
                             __ATOMIC_RELEASE, __HIP_MEMORY_SCOPE_AGENT);
        } else {                                 // wg0: fold top 3 nodes
          while (__hip_atomic_load(&flags[1], __ATOMIC_ACQUIRE,
                                   __HIP_MEMORY_SCOPE_AGENT) < t + 1) {}
          while (__hip_atomic_load(&flags[2], __ATOMIC_ACQUIRE,
                                   __HIP_MEMORY_SCOPE_AGENT) < t + 1) {}
          while (__hip_atomic_load(&flags[3], __ATOMIC_ACQUIRE,
                                   __HIP_MEMORY_SCOPE_AGENT) < t + 1) {}
          const float q4 = __hip_atomic_load(&qslot[(t & 1) * NWG + 1],
                               __ATOMIC_RELAXED, __HIP_MEMORY_SCOPE_AGENT);
          const float q5 = __hip_atomic_load(&qslot[(t & 1) * NWG + 2],
                               __ATOMIC_RELAXED, __HIP_MEMORY_SCOPE_AGENT);
          const float q6 = __hip_atomic_load(&qslot[(t & 1) * NWG + 3],
                               __ATOMIC_RELAXED, __HIP_MEMORY_SCOPE_AGENT);
          const float O1 = mc_step(p1, lat[1] + qsub + q4, I1o, O1o, dtf);
          const float O2 = mc_step(p2, lat[2] + q5 + q6,   I2o, O2o, dtf);
          const float O0 = mc_step(p0, lat[0] + O1 + O2,   I0o, O0o, dtf);
          out[t] = O0;                           // outlet discharge
          __hip_atomic_store(gdone, t + 1,
                             __ATOMIC_RELEASE, __HIP_MEMORY_SCOPE_AGENT);
        }
        // slack-1 throttle: keep double-buffered qslot parity safe
        while (__hip_atomic_load(gdone, __ATOMIC_ACQUIRE,
                                 __HIP_MEMORY_SCOPE_AGENT) < t) {}
      }
#if HAVE_TDM
      __builtin_amdgcn_s_wait_tensorcnt(0);      // next-row DMA landed
#endif
    }
    __syncthreads();
#if HAVE_CBAR
    if (wv == 0) __builtin_amdgcn_s_cluster_barrier();  // NOP if not clustered
    __syncthreads();
#endif
  }
}

extern "C" void kernel_launch(void* const* d_in, const int* in_sizes, int n_in,
                              void* d_out, int out_size, void* d_ws, size_t ws_size,
                              hipStream_t stream) {
  (void)in_sizes; (void)n_in; (void)out_size; (void)ws_size;
  const float* lat = (const float*)d_in[0];
  const float* mn  = (const float*)d_in[1];
  const float* len = (const float*)d_in[2];
  const float* slp = (const float*)d_in[3];
  const float* wcf = (const float*)d_in[4];
  const float* wex = (const float*)d_in[5];
  const float* dcf = (const float*)d_in[6];
  const float* dex = (const float*)d_in[7];
  const int*   dtp = (const int*)d_in[8];
  float* out = (float*)d_out;
  int*   ws  = (int*)d_ws;

  const int smem_bytes =
      (2 * LATSTRIDE + 2048 + 5 * NPARAM) * (int)sizeof(float);   // ~92 KB
  (void)hipFuncSetAttribute((const void*)mc_route_kernel,
                            hipFuncAttributeMaxDynamicSharedMemorySize, smem_bytes);

  mc_zero_ws<<<1, 32, 0, stream>>>(ws);
  mc_route_kernel<<<NWG, TPB, smem_bytes, stream>>>(
      lat, mn, len, slp, wcf, wex, dcf, dex, dtp, out, ws);
}